// PerfectMemorySSM_25443386262310
// MI455X (gfx1250) — compile-verified
//
#include <hip/hip_runtime.h>

typedef __attribute__((ext_vector_type(2))) float v2f;
typedef __attribute__((ext_vector_type(8))) float v8f;

#define HIDDEN 16

// ---------------------------------------------------------------------------
// Phase 1: memory[b,k] = sum_t sigmoid(x_t . Wg + bg) * tanh(x_t @ Wu + bu)[k]
// One block per batch row; 256 threads stride over T; deterministic LDS tree
// reduction (fixed order -> bitwise-stable across replays).
// ---------------------------------------------------------------------------
__global__ void __launch_bounds__(256) ssm_phase1(
    const int* __restrict__ seq, const float* __restrict__ embed,
    const float* __restrict__ Wg, const float* __restrict__ bg,
    const float* __restrict__ Wu, const float* __restrict__ bu,
    float* __restrict__ mem, int T)
{
    __shared__ float sWu[HIDDEN * HIDDEN];
    __shared__ float sWg[HIDDEN];
    __shared__ float sbu[HIDDEN];
    __shared__ float red[HIDDEN * 256];   // 16 KB

    const int b   = blockIdx.x;
    const int tid = threadIdx.x;

    if (tid < HIDDEN * HIDDEN) sWu[tid] = Wu[tid];
    if (tid < HIDDEN) { sWg[tid] = Wg[tid]; sbu[tid] = bu[tid]; }
    __syncthreads();

    const float bgv = bg[0];

    float acc[HIDDEN];
#pragma unroll
    for (int k = 0; k < HIDDEN; ++k) acc[k] = 0.0f;

    const int* srow = seq + (long)b * T;
    for (int t = tid; t < T; t += 256) {
        const int idx = srow[t];
        const float4* xp = (const float4*)(embed + (long)idx * HIDDEN);
        float x[HIDDEN];
#pragma unroll
        for (int i = 0; i < 4; ++i) {
            float4 v = xp[i];
            x[4*i+0] = v.x; x[4*i+1] = v.y; x[4*i+2] = v.z; x[4*i+3] = v.w;
        }
        float gd = bgv;
#pragma unroll
        for (int h = 0; h < HIDDEN; ++h) gd += x[h] * sWg[h];
        const float g = 1.0f / (1.0f + __expf(-gd));
#pragma unroll
        for (int k = 0; k < HIDDEN; ++k) {
            float u = sbu[k];
#pragma unroll
            for (int h = 0; h < HIDDEN; ++h) u += x[h] * sWu[h * HIDDEN + k];
            acc[k] += g * tanhf(u);
        }
    }

#pragma unroll
    for (int k = 0; k < HIDDEN; ++k) red[k * 256 + tid] = acc[k];
    for (int s = 128; s > 0; s >>= 1) {
        __syncthreads();
        if (tid < s) {
#pragma unroll
            for (int k = 0; k < HIDDEN; ++k)
                red[k * 256 + tid] += red[k * 256 + tid + s];
        }
    }
    __syncthreads();
    if (tid < HIDDEN) mem[b * HIDDEN + tid] = red[tid * 256];
}

// ---------------------------------------------------------------------------
// Phase 2: out[256, vocab] = memory[256,16] @ Wo[16,vocab] + bo
// 8 waves per block, one 16x16 output tile per wave.
// K=16 via 4 chained V_WMMA_F32_16X16X4_F32 (exact f32 math).
// ISA 7.12.2 f32 layouts:
//   A 16x4 : lane l -> M = l%16, VGPR j holds K = j + 2*(l/16)
//   B 4x16 : lane l -> N = l%16, VGPR j holds K = j + 2*(l/16)
//   C/D    : lane l -> N = l%16, VGPR j holds M = j + 8*(l/16)
// CVOCAB > 0 bakes the vocab stride into immediates: B-load offsets
// (max 13*vocab*4 ~ 2.6 MB) and store row offsets (max 7*vocab*4 ~ 1.4 MB)
// both fit the signed 24-bit instruction IOFFSET, so the compiler folds the
// whole tile into one base address + immediate-offset loads/stores.
// Tail handling: clamp load column to vocab-1 (invalid columns feed only
// unstored outputs) -> loads and WMMA branch-free with full EXEC.
// ---------------------------------------------------------------------------
template <int CVOCAB>
__global__ void __launch_bounds__(256) ssm_phase2(
    const float* __restrict__ mem, const float* __restrict__ Wo,
    const float* __restrict__ bo, float* __restrict__ out,
    int vocab_rt, int ntiles)
{
    const int vocab = (CVOCAB > 0) ? CVOCAB : vocab_rt;

    const int lane  = threadIdx.x & 31;
    const int wave  = threadIdx.x >> 5;
    const int mtile = blockIdx.x;                 // 16 tiles over B=256
    const int ntile = blockIdx.y * 8 + wave;
    if (ntile >= ntiles) return;                  // wave-uniform exit

    const int col  = lane & 15;                   // N (or M for A) within tile
    const int half = lane >> 4;                   // 0 or 1
    const int n    = ntile * 16 + col;
    const int nc   = (n < vocab) ? n : (vocab - 1);   // clamped load column

    // A fragments: memory tile rows (M = batch), K sliced in chunks of 4.
    const int abase = (mtile * 16 + col) * HIDDEN;
    v2f a[4];
#pragma unroll
    for (int kb = 0; kb < 4; ++kb) {
        const int k = kb * 4 + 2 * half;
        a[kb].x = mem[abase + k];
        a[kb].y = mem[abase + k + 1];
    }

    // B fragments: Wo tile (K x N), column nc per lane; one base pointer,
    // k*vocab strides become immediate offsets when CVOCAB > 0.
    const float* bbase = Wo + 2 * half * vocab + nc;
    v2f bf[4];
#pragma unroll
    for (int kb = 0; kb < 4; ++kb) {
        bf[kb].x = bbase[(kb * 4)     * vocab];
        bf[kb].y = bbase[(kb * 4 + 1) * vocab];
    }

    // C init: bias depends only on N (= lane column), identical in all 8 VGPRs
    const float bias = bo[nc];
    v8f acc;
#pragma unroll
    for (int j = 0; j < 8; ++j) acc[j] = bias;

#pragma unroll
    for (int kb = 0; kb < 4; ++kb)
        acc = __builtin_amdgcn_wmma_f32_16x16x4_f32(
            /*neg_a=*/false, a[kb], /*neg_b=*/false, bf[kb],
            /*c_mod=*/(short)0, acc, /*reuse_a=*/false, /*reuse_b=*/false);

    // Store: one per-lane base; j*vocab row strides fold into immediates
    // when CVOCAB > 0. Row-half segments are 64B contiguous per store.
    float* sbase = out + (size_t)(mtile * 16 + half * 8) * vocab + n;
    if (n < vocab) {
#pragma unroll
        for (int j = 0; j < 8; ++j)
            sbase[j * vocab] = acc[j];
    }
}

// ---------------------------------------------------------------------------
extern "C" void kernel_launch(void* const* d_in, const int* in_sizes, int n_in,
                              void* d_out, int out_size, void* d_ws, size_t ws_size,
                              hipStream_t stream) {
    const int*   seq   = (const int*)d_in[0];
    const float* embed = (const float*)d_in[1];
    const float* Wg    = (const float*)d_in[2];
    const float* bg    = (const float*)d_in[3];
    const float* Wu    = (const float*)d_in[4];
    const float* bu    = (const float*)d_in[5];
    const float* Wo    = (const float*)d_in[6];
    const float* bo    = (const float*)d_in[7];

    const int vocab = in_sizes[7];            // 50257 (bo length)
    const int B     = out_size / vocab;       // 256
    const int T     = in_sizes[0] / B;        // 8192

    float* mem = (float*)d_ws;                // B*16 floats = 16 KB scratch

    ssm_phase1<<<B, 256, 0, stream>>>(seq, embed, Wg, bg, Wu, bu, mem, T);

    const int ntiles = (vocab + 15) / 16;     // 3142
    dim3 grid((B + 15) / 16, (ntiles + 7) / 8);
    float* outf = (float*)d_out;
    if (vocab == 50257) {
        ssm_phase2<50257><<<grid, 256, 0, stream>>>(mem, Wo, bo, outf, vocab, ntiles);
    } else {
        ssm_phase2<0><<<grid, 256, 0, stream>>>(mem, Wo, bo, outf, vocab, ntiles);
    }
}